// GIN_19404662243721
// MI455X (gfx1250) — compile-verified
//
#include <hip/hip_runtime.h>

typedef __attribute__((ext_vector_type(2))) float v2f;
typedef __attribute__((ext_vector_type(8))) float v8f;

#define NNODES 40000
#define CH 128
#define NCLS 40
#define NCLSP 48        // classifier cols padded to 3 WMMA tiles
#define H1_STRIDE 132   // padded LDS row stride (floats) to spread banks

// ---------------------------------------------------------------------------
// Zero-fill (float4 granularity)
// ---------------------------------------------------------------------------
__global__ void zero_f4(float4* __restrict__ p, int n4) {
  int i = blockIdx.x * blockDim.x + threadIdx.x;
  if (i < n4) p[i] = make_float4(0.f, 0.f, 0.f, 0.f);
}

// ---------------------------------------------------------------------------
// Weight repack for V_WMMA_F32_16X16X4_F32 B-fragments.
// B fragment needs, per lane: {W[kk][n], W[kk+1][n]} with kk = 4g + 2h
// (h = lane>>4). Store those pairs contiguously:
//   Wp[(g*2 + h) * ncols_out + n] = {W[4g+2h][n], W[4g+2h+1][n]}
// so the GEMM inner loop is one coalesced global_load_b64 per tile.
// Pads n >= ncols_in with zeros (classifier: 40 -> 48).
// ---------------------------------------------------------------------------
__global__ void pack_w(const float* __restrict__ W, v2f* __restrict__ Wp,
                       int ncols_in, int ncols_out) {
  int idx = blockIdx.x * blockDim.x + threadIdx.x;
  int total = 64 * ncols_out;               // 32 k-groups * 2 halves
  if (idx >= total) return;
  int g2 = idx / ncols_out;
  int n  = idx % ncols_out;
  int k  = (g2 >> 1) * 4 + (g2 & 1) * 2;
  v2f v;
  v.x = (n < ncols_in) ? W[(size_t)k * ncols_in + n]       : 0.f;
  v.y = (n < ncols_in) ? W[(size_t)(k + 1) * ncols_in + n] : 0.f;
  Wp[idx] = v;
}

// ---------------------------------------------------------------------------
// Scatter-add aggregation: one wave32 per edge, one float4 per lane.
// Gather h[src]-row (coalesced 512B per wave) and atomically accumulate into
// agg[dst]-row. All arrays are L2-resident (192MB L2); f32-add atomics resolve
// in the L2 atomic units.
// ---------------------------------------------------------------------------
__global__ void scatter_add(const float* __restrict__ h,
                            const int* __restrict__ src,
                            const int* __restrict__ dst,
                            float* __restrict__ agg, int E) {
  int gid  = blockIdx.x * blockDim.x + threadIdx.x;
  int e    = gid >> 5;
  int lane = gid & 31;
  if (e >= E) return;
  int s = src[e];
  int d = dst[e];
  const float4 v = *(const float4*)(h + (size_t)s * CH + lane * 4);
  float* o = agg + (size_t)d * CH + lane * 4;
  atomicAdd(o + 0, v.x);
  atomicAdd(o + 1, v.y);
  atomicAdd(o + 2, v.z);
  atomicAdd(o + 3, v.w);
}

// ---------------------------------------------------------------------------
// Fused GIN MLP: out = relu( relu((X+AGG) @ Wa + ba) @ Wb + bb )
// Block = 256 threads (8 waves) handles 128 rows; wave w owns rows
// [block*128 + 16w, +16). GEMMs use V_WMMA_F32_16X16X4_F32.
//
// Fragment layouts (CDNA5 ISA 7.12.2, 32-bit):
//   A 16x4 : lanes 0-15 -> row=lane, K = k0,k0+1 ; lanes 16-31 -> K = k0+2,k0+3
//   B 4x16 : lanes 0-15 -> col=lane, same K split  (served by packed weights)
//   C/D    : VGPR r, lanes 0-15 -> M=r, lanes 16-31 -> M=r+8 ; N = lane&15
//
// Out-of-range rows use a clamped (valid) row index instead of EXEC-divergent
// zero fill; their outputs are masked at the store. H1 round-trips through a
// wave-private LDS region (no barrier needed; same-wave LDS RAW ordering is
// enforced by compiler-inserted s_wait_dscnt).
// ---------------------------------------------------------------------------
__global__ void __launch_bounds__(256)
gin_mlp(const float* __restrict__ X, const float* __restrict__ AGG,
        const v2f* __restrict__ Wap, const float* __restrict__ ba,
        const v2f* __restrict__ Wbp, const float* __restrict__ bb,
        float* __restrict__ out) {
  extern __shared__ float lds[];
  const int lane  = threadIdx.x & 31;
  const int wave  = threadIdx.x >> 5;
  float* h1       = lds + wave * 16 * H1_STRIDE;
  const int m0    = blockIdx.x * 128 + wave * 16;
  const int mr    = lane & 15;          // row within 16-row tile
  const int nl    = lane & 15;          // col within 16-col tile
  const int half  = lane >> 4;          // which K half-pair this lane provides
  const int khalf = half * 2;
  const int row   = m0 + mr;
  const int arow  = (row < NNODES) ? row : 0;   // clamped: loads always valid

  const float* xrow = X   + (size_t)arow * CH;
  const float* grw  = AGG + (size_t)arow * CH;

  // ---------------- GEMM1: H1 = relu((X+AGG) @ Wa + ba) ----------------
  v8f acc[8];
  for (int t = 0; t < 8; ++t) {
    float b = ba[t * 16 + nl];
    for (int r = 0; r < 8; ++r) acc[t][r] = b;
  }
  for (int g = 0; g < 32; ++g) {
    const int kk = g * 4 + khalf;
    v2f a;
    a.x = xrow[kk]     + grw[kk];
    a.y = xrow[kk + 1] + grw[kk + 1];
    const v2f* bp = Wap + (size_t)(g * 2 + half) * CH + nl;
    for (int t = 0; t < 8; ++t)
      acc[t] = __builtin_amdgcn_wmma_f32_16x16x4_f32(
          false, a, false, bp[t * 16], (short)0, acc[t], false, false);
  }
  // ReLU, stash H1 in wave-private LDS (D-layout -> row-major)
  for (int t = 0; t < 8; ++t)
    for (int r = 0; r < 8; ++r) {
      const int lr = r + 8 * half;
      const float v = acc[t][r];
      h1[lr * H1_STRIDE + t * 16 + nl] = v > 0.f ? v : 0.f;
    }

  // ---------------- GEMM2: out = relu(H1 @ Wb + bb) ----------------
  v8f acc2[8];
  for (int t = 0; t < 8; ++t) {
    float b = bb[t * 16 + nl];
    for (int r = 0; r < 8; ++r) acc2[t][r] = b;
  }
  for (int g = 0; g < 32; ++g) {
    const int kk = g * 4 + khalf;
    v2f a;
    a.x = h1[mr * H1_STRIDE + kk];
    a.y = h1[mr * H1_STRIDE + kk + 1];
    const v2f* bp = Wbp + (size_t)(g * 2 + half) * CH + nl;
    for (int t = 0; t < 8; ++t)
      acc2[t] = __builtin_amdgcn_wmma_f32_16x16x4_f32(
          false, a, false, bp[t * 16], (short)0, acc2[t], false, false);
  }
  for (int t = 0; t < 8; ++t)
    for (int r = 0; r < 8; ++r) {
      const int lr   = r + 8 * half;
      const int orow = m0 + lr;
      if (orow < NNODES) {
        const float v = acc2[t][r];
        out[(size_t)orow * CH + t * 16 + nl] = v > 0.f ? v : 0.f;
      }
    }
}

// ---------------------------------------------------------------------------
// Classifier: out[N,40] = H @ Wlin + blin. Wlin pre-packed & padded to 48
// cols -> 3 unmasked WMMA tiles; stores masked to 40 cols.
// ---------------------------------------------------------------------------
__global__ void __launch_bounds__(256)
final_linear(const float* __restrict__ H, const v2f* __restrict__ Wp,
             const float* __restrict__ B, float* __restrict__ out) {
  const int lane  = threadIdx.x & 31;
  const int wave  = threadIdx.x >> 5;
  const int m0    = blockIdx.x * 128 + wave * 16;
  const int mr    = lane & 15;
  const int nl    = lane & 15;
  const int half  = lane >> 4;
  const int khalf = half * 2;
  const int row   = m0 + mr;
  const int arow  = (row < NNODES) ? row : 0;
  const float* hrow = H + (size_t)arow * CH;

  v8f acc[3];
  for (int t = 0; t < 3; ++t) {
    const int n = t * 16 + nl;
    const float b = (n < NCLS) ? B[n] : 0.f;
    for (int r = 0; r < 8; ++r) acc[t][r] = b;
  }
  for (int g = 0; g < 32; ++g) {
    const int kk = g * 4 + khalf;
    v2f a;
    a.x = hrow[kk];
    a.y = hrow[kk + 1];
    const v2f* bp = Wp + (size_t)(g * 2 + half) * NCLSP + nl;
    for (int t = 0; t < 3; ++t)
      acc[t] = __builtin_amdgcn_wmma_f32_16x16x4_f32(
          false, a, false, bp[t * 16], (short)0, acc[t], false, false);
  }
  for (int t = 0; t < 3; ++t)
    for (int r = 0; r < 8; ++r) {
      const int lr   = r + 8 * half;
      const int orow = m0 + lr;
      const int col  = t * 16 + nl;
      if (orow < NNODES && col < NCLS)
        out[(size_t)orow * NCLS + col] = acc[t][r];
    }
}

// ---------------------------------------------------------------------------
extern "C" void kernel_launch(void* const* d_in, const int* in_sizes, int n_in,
                              void* d_out, int out_size, void* d_ws, size_t ws_size,
                              hipStream_t stream) {
  const float* x    = (const float*)d_in[0];
  const int*   edge = (const int*)d_in[1];   // [2, E]
  const float* w1a  = (const float*)d_in[2];
  const float* b1a  = (const float*)d_in[3];
  const float* w1b  = (const float*)d_in[4];
  const float* b1b  = (const float*)d_in[5];
  const float* w2a  = (const float*)d_in[6];
  const float* b2a  = (const float*)d_in[7];
  const float* w2b  = (const float*)d_in[8];
  const float* b2b  = (const float*)d_in[9];
  const float* wlin = (const float*)d_in[10];
  const float* blin = (const float*)d_in[11];
  float* out = (float*)d_out;

  const int E = in_sizes[1] / 2;
  const int* src = edge;
  const int* dst = edge + E;

  // Workspace layout (floats): 3 feature buffers + packed weights
  float* agg = (float*)d_ws;                 // N*CH
  float* h1  = agg + (size_t)NNODES * CH;    // N*CH
  float* h2  = h1  + (size_t)NNODES * CH;    // N*CH
  v2f* w1ap  = (v2f*)(h2 + (size_t)NNODES * CH);   // 64*CH v2f each
  v2f* w1bp  = w1ap + 64 * CH;
  v2f* w2ap  = w1bp + 64 * CH;
  v2f* w2bp  = w2ap + 64 * CH;
  v2f* wlp   = w2bp + 64 * CH;                     // 64*NCLSP v2f

  const int nf4 = NNODES * CH / 4;
  dim3 blk(256);
  dim3 zgrid((nf4 + 255) / 256);
  dim3 sgrid(((size_t)E * 32 + 255) / 256);
  dim3 mgrid((NNODES + 127) / 128);
  dim3 pgrid((64 * CH + 255) / 256);
  dim3 pgridl((64 * NCLSP + 255) / 256);
  size_t smem = 8 * 16 * H1_STRIDE * sizeof(float);  // 67.6 KB of 320 KB WGP LDS

  // One-shot weight repacks (deterministic, graph-capture safe)
  pack_w<<<pgrid,  blk, 0, stream>>>(w1a,  w1ap, CH,   CH);
  pack_w<<<pgrid,  blk, 0, stream>>>(w1b,  w1bp, CH,   CH);
  pack_w<<<pgrid,  blk, 0, stream>>>(w2a,  w2ap, CH,   CH);
  pack_w<<<pgrid,  blk, 0, stream>>>(w2b,  w2bp, CH,   CH);
  pack_w<<<pgridl, blk, 0, stream>>>(wlin, wlp,  NCLS, NCLSP);

  // Layer 1
  zero_f4    <<<zgrid, blk, 0, stream>>>((float4*)agg, nf4);
  scatter_add<<<sgrid, blk, 0, stream>>>(x, src, dst, agg, E);
  gin_mlp    <<<mgrid, blk, smem, stream>>>(x, agg, w1ap, b1a, w1bp, b1b, h1);
  // Layer 2
  zero_f4    <<<zgrid, blk, 0, stream>>>((float4*)agg, nf4);
  scatter_add<<<sgrid, blk, 0, stream>>>(h1, src, dst, agg, E);
  gin_mlp    <<<mgrid, blk, smem, stream>>>(h1, agg, w2ap, b2a, w2bp, b2b, h2);
  // Classifier
  final_linear<<<mgrid, blk, 0, stream>>>(h2, wlp, blin, out);
}